// Similarity_Search_76484777607595
// MI455X (gfx1250) — compile-verified
//
#include <hip/hip_runtime.h>

typedef float v2f __attribute__((ext_vector_type(2)));
typedef float v8f __attribute__((ext_vector_type(8)));
typedef int   v4i __attribute__((ext_vector_type(4)));

#define NQ        512      // queries (D)
#define CDIM      1024     // feature dim (C)
#define NROWS     200000   // db rows (N)
#define DBSTRIDE  1025     // db row stride in floats (features + id)
#define TOPK      10
#define MIN_SIM   0.8f

#define MB        32       // queries per block (2 WMMA M tiles)
#define NB        32       // db rows per wave group (2 WMMA N tiles)
#define KB        16       // K-slice staged in LDS
#define KBP       20       // padded LDS row stride; KBP%64==20, gcd=4 -> conflict-free b64 reads
#define NWAVES    8
#define CHUNKS    25
#define CHUNK_ROWS 8000    // NROWS / CHUNKS
#define GROUPS_PER_CHUNK 250  // CHUNK_ROWS / NB

// ---- CDNA5 async global->LDS staging (ASYNCcnt path), with fallback ----
#if __has_builtin(__builtin_amdgcn_global_load_async_to_lds_b32) && \
    __has_builtin(__builtin_amdgcn_global_load_async_to_lds_b128)
#define USE_ASYNC_LDS 1
#else
#define USE_ASYNC_LDS 0
#endif

#if __has_builtin(__builtin_amdgcn_s_wait_asynccnt)
#define WAIT_ASYNC0() __builtin_amdgcn_s_wait_asynccnt(0)
#else
#define WAIT_ASYNC0() asm volatile("s_wait_asynccnt 0" ::: "memory")
#endif

#if __has_builtin(__builtin_amdgcn_s_wait_dscnt)
#define WAIT_DS0() __builtin_amdgcn_s_wait_dscnt(0)
#else
#define WAIT_DS0() asm volatile("s_wait_dscnt 0" ::: "memory")
#endif

#if USE_ASYNC_LDS
// Builtin signatures (from hipcc diagnostics):
//   b32 : (AS(1) int*,  AS(3) int*,  imm offset, imm cpol)
//   b128: (AS(1) int4*, AS(3) int4*, imm offset, imm cpol)
typedef __attribute__((address_space(1))) int  gas_i32;
typedef __attribute__((address_space(3))) int  las_i32;
typedef __attribute__((address_space(1))) v4i  gas_v4i;
typedef __attribute__((address_space(3))) v4i  las_v4i;

__device__ __forceinline__ void async_ld_b32(const float* g, float* l) {
    __builtin_amdgcn_global_load_async_to_lds_b32((gas_i32*)g, (las_i32*)l, 0, 0);
}
__device__ __forceinline__ void async_ld_b128(const float* g, float* l) {
    __builtin_amdgcn_global_load_async_to_lds_b128((gas_v4i*)g, (las_v4i*)l, 0, 0);
}
#endif

__device__ __forceinline__ v8f wmma_f32_4(v2f a, v2f b, v8f c) {
    // D = A(16x4 f32) * B(4x16 f32) + C(16x16 f32)
    return __builtin_amdgcn_wmma_f32_16x16x4_f32(
        /*neg_a=*/false, a, /*neg_b=*/false, b,
        /*c_mod=*/(short)0, c, /*reuse_a=*/false, /*reuse_b=*/false);
}

// Insert one 16x16 sim tile (WMMA C layout) into this wave's per-query top-10
// lists in LDS. Lane L holds: query = qoff + v + 8*(L>>4), column =
// colBase + (L&15), value = acc[v]. Fast path: one LDS threshold read +
// ballot; rare path: serialized sorted insertion (one lane per half-group).
__device__ __forceinline__ void topk_insert_tile(
    v8f acc, int qoff, int colBase, float (*tv)[TOPK], int (*ti)[TOPK], int lane)
{
    const int m = lane & 15;
    const int h = lane >> 4;
    const int col = colBase + m;
#pragma unroll
    for (int v = 0; v < 8; ++v) {
        const int q = qoff + v + 8 * h;
        const float val = acc[v];
        const float thr = tv[q][TOPK - 1];
        unsigned long long bal = __ballot(val > thr);
        unsigned mymask = (h == 0) ? (unsigned)(bal & 0xFFFFull)
                                   : (unsigned)((bal >> 16) & 0xFFFFull);
        while (mymask) {
            const int srcm = __ffs(mymask) - 1;   // 0..15 within half-group
            mymask &= (mymask - 1);
            if (m == srcm) {
                if (val > tv[q][TOPK - 1]) {      // re-check: bar may have risen
                    int p = TOPK - 1;
                    while (p > 0 && tv[q][p - 1] < val) {
                        tv[q][p] = tv[q][p - 1];
                        ti[q][p] = ti[q][p - 1];
                        --p;
                    }
                    tv[q][p] = val;
                    ti[q][p] = col;
                }
            }
        }
    }
}

__global__ __launch_bounds__(256)
void simsearch_topk_kernel(const float* __restrict__ desc,
                           const float* __restrict__ db,
                           float* __restrict__ wsV,
                           int* __restrict__ wsI)
{
    __shared__ float sA[NWAVES][MB * KBP];   // per-wave A slice: 32 x 16 (padded)
    __shared__ float sB[NWAVES][NB * KBP];   // per-wave B slice: 32 x 16 (padded)
    __shared__ float tv[NWAVES][MB][TOPK];
    __shared__ int   ti[NWAVES][MB][TOPK];

    const int tid  = threadIdx.x;
    const int wave = tid >> 5;
    const int lane = tid & 31;
    const int m    = lane & 15;
    const int h    = lane >> 4;
    const int qbase    = blockIdx.x * MB;
    const int chunk    = blockIdx.y;
    const int rowBase0 = chunk * CHUNK_ROWS;

    // Init this wave's top-k lists (wave-private; same-wave DS ops in-order).
    for (int j = lane; j < MB * TOPK; j += 32) {
        tv[wave][j / TOPK][j % TOPK] = -1e30f;
        ti[wave][j / TOPK][j % TOPK] = -1;
    }

    float* A = sA[wave];
    float* B = sB[wave];

    // Per-lane staging bases (loop-invariant pieces hoisted).
    const float* gA = desc + (size_t)(qbase + (lane >> 2)) * CDIM + (lane & 3) * 4;
    float*       lA = A + (lane >> 2) * KBP + (lane & 3) * 4;   // 16B-aligned
    const int    brow = lane >> 4;          // 0..1
    const int    bcol = lane & 15;
    float*       lB = B + brow * KBP + bcol;

    for (int G = wave; G < GROUPS_PER_CHUNK; G += NWAVES) {
        const int rowBase = rowBase0 + G * NB;
        const float* gB = db + (size_t)(rowBase + brow) * DBSTRIDE + bcol;

        v8f acc00 = {}, acc01 = {}, acc10 = {}, acc11 = {};

        for (int ko = 0; ko < CDIM; ko += KB) {
#if USE_ASYNC_LDS
            // Protect WAR on the wave-private slice buffers: make sure the
            // previous slice's indexed DS reads are complete before async
            // writes can land (async LDS writes are unordered vs DS ops).
            WAIT_DS0();
            // Stage A: 32 rows x 16 cols, one b128 per lane per instruction.
#pragma unroll
            for (int i = 0; i < 4; ++i)
                async_ld_b128(gA + ko + (size_t)i * 8 * CDIM, lA + i * 8 * KBP);
            // Stage B: 32 rows x 16 cols, b32 (db rows only 4B-aligned).
#pragma unroll
            for (int i = 0; i < 16; ++i)
                async_ld_b32(gB + ko + (size_t)i * 2 * DBSTRIDE, lB + i * 2 * KBP);
            WAIT_ASYNC0();
#else
            // Fallback: explicit VGPR staging.
#pragma unroll
            for (int i = 0; i < (MB * KB) / 32; ++i) {
                const int e = i * 32 + lane;
                const int r = e >> 4, c = e & 15;
                A[r * KBP + c] = desc[(size_t)(qbase + r) * CDIM + ko + c];
            }
#pragma unroll
            for (int i = 0; i < (NB * KB) / 32; ++i) {
                const int e = i * 32 + lane;
                const int r = e >> 4, c = e & 15;
                B[r * KBP + c] = db[(size_t)(rowBase + r) * DBSTRIDE + ko + c];
            }
#endif
            // 2x2 register tile: each staged B element feeds both M tiles.
#pragma unroll
            for (int kk = 0; kk < KB; kk += 4) {
                const int fo = kk + 2 * h;
                v2f a0 = *(const v2f*)(A + m * KBP + fo);
                v2f a1 = *(const v2f*)(A + (16 + m) * KBP + fo);
                v2f b0 = *(const v2f*)(B + m * KBP + fo);
                v2f b1 = *(const v2f*)(B + (16 + m) * KBP + fo);
                acc00 = wmma_f32_4(a0, b0, acc00);
                acc01 = wmma_f32_4(a0, b1, acc01);
                acc10 = wmma_f32_4(a1, b0, acc10);
                acc11 = wmma_f32_4(a1, b1, acc11);
            }
        }
        topk_insert_tile(acc00,  0, rowBase +  0, tv[wave], ti[wave], lane);
        topk_insert_tile(acc01,  0, rowBase + 16, tv[wave], ti[wave], lane);
        topk_insert_tile(acc10, 16, rowBase +  0, tv[wave], ti[wave], lane);
        topk_insert_tile(acc11, 16, rowBase + 16, tv[wave], ti[wave], lane);
    }

    __syncthreads();

    // Merge the 8 per-wave lists into one top-10 per query; write workspace.
    if (tid < MB) {
        float mv[TOPK];
        int   mi[TOPK];
#pragma unroll
        for (int s = 0; s < TOPK; ++s) { mv[s] = -1e30f; mi[s] = -1; }
        for (int w = 0; w < NWAVES; ++w) {
            for (int s = 0; s < TOPK; ++s) {
                const float v = tv[w][tid][s];
                const int   x = ti[w][tid][s];
                if (v > mv[TOPK - 1]) {
                    int p = TOPK - 1;
                    while (p > 0 && mv[p - 1] < v) {
                        mv[p] = mv[p - 1]; mi[p] = mi[p - 1]; --p;
                    }
                    mv[p] = v; mi[p] = x;
                }
            }
        }
        const int q = qbase + tid;
        const int base = (q * CHUNKS + chunk) * TOPK;
        for (int s = 0; s < TOPK; ++s) {
            wsV[base + s] = mv[s];
            wsI[base + s] = mi[s];
        }
    }
}

__global__ __launch_bounds__(256)
void simsearch_finalize_kernel(const float* __restrict__ boxes,
                               const float* __restrict__ db,
                               const float* __restrict__ wsV,
                               const int* __restrict__ wsI,
                               float* __restrict__ out)
{
    const int q = blockIdx.x * blockDim.x + threadIdx.x;
    if (q >= NQ) return;

    // final_boxes passthrough
#pragma unroll
    for (int j = 0; j < 4; ++j) out[q * 4 + j] = boxes[q * 4 + j];

    // Merge 25 partial top-10 lists into the global top-10 for this query.
    float bv[TOPK];
    int   bi[TOPK];
#pragma unroll
    for (int s = 0; s < TOPK; ++s) { bv[s] = -1e30f; bi[s] = -1; }
    for (int c = 0; c < CHUNKS; ++c) {
        const int base = (q * CHUNKS + c) * TOPK;
        for (int s = 0; s < TOPK; ++s) {
            const float v = wsV[base + s];
            const int   x = wsI[base + s];
            if (v > bv[TOPK - 1]) {
                int p = TOPK - 1;
                while (p > 0 && bv[p - 1] < v) {
                    bv[p] = bv[p - 1]; bi[p] = bi[p - 1]; --p;
                }
                bv[p] = v; bi[p] = x;
            }
        }
    }

    // Mask at sim >= MIN_SIM (reference mask_sim); fetch class ids.
    int  ids[TOPK];
    bool ok[TOPK];
    for (int s = 0; s < TOPK; ++s) {
        ok[s]  = (bi[s] >= 0) && (bv[s] >= MIN_SIM);
        ids[s] = ok[s] ? (int)db[(size_t)bi[s] * DBSTRIDE + CDIM] : -1;
    }

    // Majority vote over <=10 entries; argmax ties -> smallest class id
    // (jnp.argmax over the dense class axis returns the first index).
    int bestCount = 0;
    int bestId = 0x7fffffff;
    for (int s = 0; s < TOPK; ++s) {
        if (!ok[s]) continue;
        const int id = ids[s];
        int cnt = 0;
        for (int t = 0; t < TOPK; ++t) cnt += (ok[t] && ids[t] == id) ? 1 : 0;
        if (cnt > bestCount || (cnt == bestCount && id < bestId)) {
            bestCount = cnt;
            bestId = id;
        }
    }
    const int result = (bestCount > 0) ? bestId : -1;

    float score = 0.0f;
    for (int s = 0; s < TOPK; ++s)
        if (ok[s] && ids[s] == result) score = fmaxf(score, bv[s]);

    out[2048 + q] = score;              // sim_scores
    out[2048 + NQ + q] = (float)result; // results
}

extern "C" void kernel_launch(void* const* d_in, const int* in_sizes, int n_in,
                              void* d_out, int out_size, void* d_ws, size_t ws_size,
                              hipStream_t stream) {
    const float* boxes = (const float*)d_in[0];   // (512, 4)
    const float* desc  = (const float*)d_in[1];   // (512, 1024)
    const float* db    = (const float*)d_in[2];   // (200000, 1025)
    float* out = (float*)d_out;                   // 2048 + 512 + 512

    float* wsV = (float*)d_ws;                                    // 512*25*10 f32
    int*   wsI = (int*)((char*)d_ws + (size_t)NQ * CHUNKS * TOPK * sizeof(float));

    dim3 gridA(NQ / MB, CHUNKS);   // (16, 25): query tiles iterate fastest so
                                   // all of them hit the same L2-resident chunk.
    simsearch_topk_kernel<<<gridA, 256, 0, stream>>>(desc, db, wsV, wsI);
    simsearch_finalize_kernel<<<dim3(2), 256, 0, stream>>>(boxes, db, wsV, wsI, out);
}